// TimestepNorm_18425409700267
// MI455X (gfx1250) — compile-verified
//
#include <hip/hip_runtime.h>
#include <hip/hip_bf16.h>
#include <stdint.h>

// TimestepNorm fused kernel for MI455X (gfx1250, wave32).
// One block per (b,g); sequence tiled into LDS with double-buffered
// gfx1250 async global->LDS loads (ASYNCcnt), parallel prefix scans for the
// running mean/var, normalize straight out of LDS. x read once, y written
// once => ~268 MB traffic => ~11.5us roofline @ 23.3 TB/s.

#define B_   4
#define L_   4096
#define D_   2048
#define G_   32
#define GS_  64
#define T_   256          // timesteps per tile
#define NT_  (L_ / T_)    // 16 tiles
#define NTH  256

__device__ __forceinline__ void async_load16(unsigned lds_byte, const void* gptr) {
  unsigned long long ga = (unsigned long long)gptr;
  // gfx1250: per-lane 16B global -> LDS, tracked by ASYNCcnt
  asm volatile("global_load_async_to_lds_b128 %0, %1, off"
               :: "v"(lds_byte), "v"(ga) : "memory");
}
__device__ __forceinline__ void wait_async_le16() {
  asm volatile("s_wait_asynccnt 0x10" ::: "memory");
}
__device__ __forceinline__ void wait_async_0() {
  asm volatile("s_wait_asynccnt 0x0" ::: "memory");
}

__global__ __launch_bounds__(NTH)
void tsnorm_kernel(const float* __restrict__ x,
                   const float* __restrict__ weight,
                   const float* __restrict__ bias,
                   const unsigned char* __restrict__ mask,
                   float* __restrict__ out) {
  extern __shared__ __align__(16) float smem[];
  float* xb0   = smem;                 // 256*64 floats = 64KB
  float* xb1   = smem + T_ * GS_;      // 64KB
  float* gm_s  = smem + 2 * T_ * GS_;  // 256
  float* meanA = gm_s  + T_;           // 256
  float* istdA = meanA + T_;           // 256
  float* vtot  = istdA + T_;           // 8
  float* gvtot = vtot  + 8;            // 8
  float* dttot = gvtot + 8;            // 8
  float* carry = dttot + 8;            // 8

  const int tid  = threadIdx.x;
  const int lane = tid & 31;
  const int wv   = tid >> 5;   // wave id 0..7
  const int sub  = tid & 15;   // element chunk (float4) within group
  const int rowq = tid >> 4;   // timestep row 0..15 per pass

  const int bIdx = blockIdx.x / G_;
  const int g    = blockIdx.x % G_;

  const size_t xBase = (size_t)bIdx * L_ * D_ + (size_t)g * GS_;
  const float* xg = x + xBase;
  float* yg = out + xBase;
  const unsigned char* mrow = mask + (size_t)bIdx * L_;

  // per-thread scale/shift for its 4 elements of the group
  float4 wv4 = ((const float4*)weight)[g * (GS_ / 4) + sub];
  float4 bv4 = ((const float4*)bias)[g * (GS_ / 4) + sub];
  const float4 sc = make_float4(wv4.x + 1.f, wv4.y + 1.f, wv4.z + 1.f, wv4.w + 1.f);

  unsigned xb_byte[2];
  xb_byte[0] = (unsigned)(unsigned long long)(void*)xb0;  // flat->LDS offset (low 32 bits)
  xb_byte[1] = (unsigned)(unsigned long long)(void*)xb1;

  auto issue_tile = [&](int tile) {
    const float* src = xg + (size_t)(tile * T_) * D_;
    unsigned lbase = xb_byte[tile & 1];
#pragma unroll
    for (int p = 0; p < 16; ++p) {
      int tt = p * 16 + rowq;
      async_load16(lbase + (unsigned)(tt * GS_ + sub * 4) * 4u,
                   src + (size_t)tt * D_ + sub * 4);
    }
  };

  issue_tile(0);
  if (NT_ > 1) issue_tile(1);

  // scan carry: prev_count=0, prev_sum=0, prev_m2=prev_var*max(prev_count,1)=1, prev_mean=0
  float countC = 0.f, sumC = 0.f, m2C = 1.f, meanC = 0.f;

  for (int i = 0; i < NT_; ++i) {
    if (i < NT_ - 1) wait_async_le16(); else wait_async_0();
    const float4* xb4 = (const float4*)((i & 1) ? xb1 : xb0);

    // ---- group means: 16 lanes x float4 per timestep, reduce via shfl_xor ----
#pragma unroll
    for (int p = 0; p < 16; ++p) {
      int tt = p * 16 + rowq;
      float4 xv = xb4[tt * (GS_ / 4) + sub];
      float s = (xv.x + xv.y) + (xv.z + xv.w);
      s += __shfl_xor(s, 1);
      s += __shfl_xor(s, 2);
      s += __shfl_xor(s, 4);
      s += __shfl_xor(s, 8);
      if (sub == 0) gm_s[tt] = s * (1.0f / GS_);
    }
    __syncthreads();

    // ---- parallel prefix scans: valid, gm*valid ----
    const int t = tid;  // one timestep per thread
    float v   = mrow[i * T_ + t] ? 1.0f : 0.0f;
    float gmv = gm_s[t];
    float sv = v, sg = gmv * v;
#pragma unroll
    for (int d2 = 1; d2 < 32; d2 <<= 1) {
      float a = __shfl_up(sv, d2);
      float c = __shfl_up(sg, d2);
      if (lane >= d2) { sv += a; sg += c; }
    }
    if (lane == 31) { vtot[wv] = sv; gvtot[wv] = sg; }
    __syncthreads();
    float vo = 0.f, go = 0.f;
    for (int w = 0; w < wv; ++w) { vo += vtot[w]; go += gvtot[w]; }
    float count_t = countC + sv + vo;
    float sum_t   = sumC   + sg + go;
    float mean_t  = sum_t / fmaxf(count_t, 1.0f);
    meanA[t] = mean_t;
    __syncthreads();

    // ---- delta-term scan (Welford M2 over group means) ----
    float mean_prev = (t == 0) ? meanC : meanA[t - 1];
    float sd = (gmv - mean_prev) * (gmv - mean_t) * v;
#pragma unroll
    for (int d2 = 1; d2 < 32; d2 <<= 1) {
      float a = __shfl_up(sd, d2);
      if (lane >= d2) sd += a;
    }
    if (lane == 31) dttot[wv] = sd;
    __syncthreads();
    float dofs = 0.f;
    for (int w = 0; w < wv; ++w) dofs += dttot[w];
    float m2_t  = m2C + sd + dofs;
    float var_t = fmaxf(m2_t / fmaxf(count_t, 1.0f), 1e-6f);
    istdA[t] = rsqrtf(var_t + 1e-5f);

    if (t == T_ - 1) {
      carry[0] = count_t; carry[1] = sum_t; carry[2] = m2_t; carry[3] = mean_t;
      if (i == NT_ - 1) {  // final timestep: tail outputs
        float* tail = out + (size_t)B_ * L_ * D_;
        if (g == 0) ((int*)tail)[bIdx] = (int)(count_t + 0.5f);       // new_count (i32)
        tail[4 + bIdx * G_ + g]            = mean_t;                  // mean_out
        tail[4 + B_ * G_ + bIdx * G_ + g]  = var_t;                   // var_out (post-floor)
      }
    }
    __syncthreads();
    countC = carry[0]; sumC = carry[1]; m2C = carry[2]; meanC = carry[3];

    // ---- normalize + store y (coalesced float4) ----
#pragma unroll
    for (int p = 0; p < 16; ++p) {
      int tt = p * 16 + rowq;
      float4 xv = xb4[tt * (GS_ / 4) + sub];
      float m  = meanA[tt];
      float is = istdA[tt];
      float4 yv;
      yv.x = (xv.x - m) * is * sc.x + bv4.x;
      yv.y = (xv.y - m) * is * sc.y + bv4.y;
      yv.z = (xv.z - m) * is * sc.z + bv4.z;
      yv.w = (xv.w - m) * is * sc.w + bv4.w;
      *((float4*)(yg + (size_t)(i * T_ + tt) * D_ + sub * 4)) = yv;
    }

    // this wave only ever touches its own x-slice, so it can refill its part
    // of the parity buffer immediately after its last read -- no barrier.
    if (i + 2 < NT_) issue_tile(i + 2);
  }
}

extern "C" void kernel_launch(void* const* d_in, const int* in_sizes, int n_in,
                              void* d_out, int out_size, void* d_ws, size_t ws_size,
                              hipStream_t stream) {
  (void)in_sizes; (void)n_in; (void)out_size; (void)d_ws; (void)ws_size;
  const float* x         = (const float*)d_in[0];
  const float* w         = (const float*)d_in[1];
  const float* b         = (const float*)d_in[2];
  const unsigned char* m = (const unsigned char*)d_in[3];  // jnp.bool_ -> 1 byte/elem
  float* out = (float*)d_out;

  const size_t smem = (size_t)(2 * T_ * GS_ + 3 * T_ + 32) * sizeof(float);  // ~131KB
  (void)hipFuncSetAttribute(reinterpret_cast<const void*>(tsnorm_kernel),
                            hipFuncAttributeMaxDynamicSharedMemorySize, (int)smem);
  tsnorm_kernel<<<dim3(B_ * G_), dim3(NTH), smem, stream>>>(x, w, b, m, out);
}